// GraphSAGE_34342558499453
// MI455X (gfx1250) — compile-verified
//
#include <hip/hip_runtime.h>

typedef __attribute__((ext_vector_type(2))) float v2f;
typedef __attribute__((ext_vector_type(8))) float v8f;

#define N_NODES 100000
#define F0 48
#define F1 64
#define F2 32

// ---------------------------------------------------------------- zero fill
__global__ void zero_f32_kernel(float4* __restrict__ p, long n4) {
    long i = (long)blockIdx.x * blockDim.x + threadIdx.x;
    long stride = (long)gridDim.x * blockDim.x;
    float4 z = make_float4(0.f, 0.f, 0.f, 0.f);
    for (; i < n4; i += stride) p[i] = z;
}

// ---------------------------------------------------------------- degree
__global__ void degree_kernel(const long long* __restrict__ dst,
                              float* __restrict__ cnt, int E) {
    int i = blockIdx.x * blockDim.x + threadIdx.x;
    int stride = gridDim.x * blockDim.x;
    for (; i < E; i += stride) {
        int d = (int)dst[i];
        unsafeAtomicAdd(&cnt[d], 1.0f);
    }
}

__global__ void invert_kernel(float* __restrict__ cnt, int n) {
    int i = blockIdx.x * blockDim.x + threadIdx.x;
    if (i < n) {
        float c = cnt[i];
        cnt[i] = (c > 0.f) ? (1.0f / c) : 0.0f;
    }
}

// ------------------------------------------------- edge scatter: x -> agg1
// 16 lanes per edge, 3 feats per lane (48 total). Gathers hit L2 (x resident).
__global__ void scatter48_kernel(const float* __restrict__ x,
                                 const long long* __restrict__ src,
                                 const long long* __restrict__ dst,
                                 float* __restrict__ agg, long E) {
    long t = (long)blockIdx.x * blockDim.x + threadIdx.x;
    long stride = (long)gridDim.x * blockDim.x;
    long total = E * 16;
    for (; t < total; t += stride) {
        long e = t >> 4;
        int f = (int)(t & 15) * 3;
        long s = src[e], d = dst[e];
        const float* xs = x + s * F0 + f;
        float* ad = agg + d * F0 + f;
        unsafeAtomicAdd(ad + 0, xs[0]);
        unsafeAtomicAdd(ad + 1, xs[1]);
        unsafeAtomicAdd(ad + 2, xs[2]);
    }
}

// ------------------------------------------------- edge scatter: h -> agg2
// 16 lanes per edge, float4 per lane (64 total).
__global__ void scatter64_kernel(const float* __restrict__ h,
                                 const long long* __restrict__ src,
                                 const long long* __restrict__ dst,
                                 float* __restrict__ agg, long E) {
    long t = (long)blockIdx.x * blockDim.x + threadIdx.x;
    long stride = (long)gridDim.x * blockDim.x;
    long total = E * 16;
    for (; t < total; t += stride) {
        long e = t >> 4;
        int f = (int)(t & 15) * 4;
        long s = src[e], d = dst[e];
        float4 v = *(const float4*)(h + s * F1 + f);
        float* ad = agg + d * F1 + f;
        unsafeAtomicAdd(ad + 0, v.x);
        unsafeAtomicAdd(ad + 1, v.y);
        unsafeAtomicAdd(ad + 2, v.z);
        unsafeAtomicAdd(ad + 3, v.w);
    }
}

// ---------------------------------------------------------- WMMA layer 1
// h = relu( mean(agg1) @ W1l^T + x @ W1r^T + b1 )   [N,48] -> [N,64]
// One wave per 16(node) x 16(out) tile; K chunked by 4 via v_wmma_f32_16x16x4_f32.
__global__ void sage_layer1_kernel(const float* __restrict__ x,
                                   const float* __restrict__ agg1,
                                   const float* __restrict__ invc,
                                   const float* __restrict__ Wl,
                                   const float* __restrict__ Wr,
                                   const float* __restrict__ bias,
                                   float* __restrict__ h) {
    int wave = (int)((blockIdx.x * blockDim.x + threadIdx.x) >> 5);
    int lane = threadIdx.x & 31;
    const int ntiles = (N_NODES / 16) * (F1 / 16);
    if (wave >= ntiles) return;            // wave-uniform: EXEC stays all-ones
    int tm = wave >> 2;                    // node tile
    int tn = wave & 3;                     // output-column tile
    int m  = lane & 15;                    // A: M index; B: N index
    int kp = (lane >> 4) * 2;              // K sub-pair: lanes 0-15 -> K 0,1; 16-31 -> K 2,3
    int arow = tm * 16 + m;                // A row (node)
    int bcol = tn * 16 + m;                // B column (output feature)
    float im = invc[arow];

    v8f c = {};
    const float* arow_p = agg1 + (long)arow * F0;
    const float* xrow_p = x    + (long)arow * F0;
    const float* wl_p   = Wl   + (long)bcol * F0;
    const float* wr_p   = Wr   + (long)bcol * F0;

    #pragma unroll
    for (int k0 = 0; k0 < F0; k0 += 4) {
        v2f a, b;
        a.x = arow_p[k0 + kp + 0] * im;
        a.y = arow_p[k0 + kp + 1] * im;
        b.x = wl_p[k0 + kp + 0];
        b.y = wl_p[k0 + kp + 1];
        c = __builtin_amdgcn_wmma_f32_16x16x4_f32(false, a, false, b,
                                                  (short)0, c, false, false);
    }
    #pragma unroll
    for (int k0 = 0; k0 < F0; k0 += 4) {
        v2f a, b;
        a.x = xrow_p[k0 + kp + 0];
        a.y = xrow_p[k0 + kp + 1];
        b.x = wr_p[k0 + kp + 0];
        b.y = wr_p[k0 + kp + 1];
        c = __builtin_amdgcn_wmma_f32_16x16x4_f32(false, a, false, b,
                                                  (short)0, c, false, false);
    }

    // D layout: VGPR r -> M = r + 8*(lane>=16), N = lane&15
    int n    = lane & 15;
    int mofs = (lane >> 4) * 8;
    float bv = bias[tn * 16 + n];
    #pragma unroll
    for (int r = 0; r < 8; ++r) {
        float v = c[r] + bv;
        v = v > 0.f ? v : 0.f;                         // ReLU
        h[(long)(tm * 16 + mofs + r) * F1 + tn * 16 + n] = v;
    }
}

// ---------------------------------------------------------- WMMA layer 2
// out = mean(agg2) @ W2l^T + h @ W2r^T + b2   [N,64] -> [N,32], no ReLU.
__global__ void sage_layer2_kernel(const float* __restrict__ h,
                                   const float* __restrict__ agg2,
                                   const float* __restrict__ invc,
                                   const float* __restrict__ Wl,
                                   const float* __restrict__ Wr,
                                   const float* __restrict__ bias,
                                   float* __restrict__ out) {
    int wave = (int)((blockIdx.x * blockDim.x + threadIdx.x) >> 5);
    int lane = threadIdx.x & 31;
    const int ntiles = (N_NODES / 16) * (F2 / 16);
    if (wave >= ntiles) return;
    int tm = wave >> 1;
    int tn = wave & 1;
    int m  = lane & 15;
    int kp = (lane >> 4) * 2;
    int arow = tm * 16 + m;
    int bcol = tn * 16 + m;
    float im = invc[arow];

    v8f c = {};
    const float* arow_p = agg2 + (long)arow * F1;
    const float* hrow_p = h    + (long)arow * F1;
    const float* wl_p   = Wl   + (long)bcol * F1;
    const float* wr_p   = Wr   + (long)bcol * F1;

    #pragma unroll
    for (int k0 = 0; k0 < F1; k0 += 4) {
        v2f a, b;
        a.x = arow_p[k0 + kp + 0] * im;
        a.y = arow_p[k0 + kp + 1] * im;
        b.x = wl_p[k0 + kp + 0];
        b.y = wl_p[k0 + kp + 1];
        c = __builtin_amdgcn_wmma_f32_16x16x4_f32(false, a, false, b,
                                                  (short)0, c, false, false);
    }
    #pragma unroll
    for (int k0 = 0; k0 < F1; k0 += 4) {
        v2f a, b;
        a.x = hrow_p[k0 + kp + 0];
        a.y = hrow_p[k0 + kp + 1];
        b.x = wr_p[k0 + kp + 0];
        b.y = wr_p[k0 + kp + 1];
        c = __builtin_amdgcn_wmma_f32_16x16x4_f32(false, a, false, b,
                                                  (short)0, c, false, false);
    }

    int n    = lane & 15;
    int mofs = (lane >> 4) * 8;
    float bv = bias[tn * 16 + n];
    #pragma unroll
    for (int r = 0; r < 8; ++r) {
        out[(long)(tm * 16 + mofs + r) * F2 + tn * 16 + n] = c[r] + bv;
    }
}

// ----------------------------------------------------------------- launch
extern "C" void kernel_launch(void* const* d_in, const int* in_sizes, int n_in,
                              void* d_out, int out_size, void* d_ws, size_t ws_size,
                              hipStream_t stream) {
    const float*     x    = (const float*)d_in[0];
    const long long* eidx = (const long long*)d_in[1];
    const float*     W1l  = (const float*)d_in[2];
    const float*     W1r  = (const float*)d_in[3];
    const float*     b1   = (const float*)d_in[4];
    const float*     W2l  = (const float*)d_in[5];
    const float*     W2r  = (const float*)d_in[6];
    const float*     b2   = (const float*)d_in[7];
    float* out = (float*)d_out;

    const long E = in_sizes[1] / 2;
    const long long* src = eidx;
    const long long* dst = eidx + E;

    // Workspace layout (floats): agg1 | agg2 | cnt/inv | h
    float* agg1 = (float*)d_ws;                 // N*48
    float* agg2 = agg1 + (long)N_NODES * F0;    // N*64
    float* cnt  = agg2 + (long)N_NODES * F1;    // N     (becomes 1/deg)
    float* h    = cnt  + N_NODES;               // N*64  (fully overwritten)

    // 1) zero accumulators + counts (contiguous N*(48+64+1) floats, %4 == 0)
    long nzero4 = (long)N_NODES * (F0 + F1 + 1) / 4;
    int zb = (int)((nzero4 + 255) / 256);
    zero_f32_kernel<<<zb, 256, 0, stream>>>((float4*)d_ws, nzero4);

    // 2) in-degree counts
    degree_kernel<<<2048, 256, 0, stream>>>(dst, cnt, (int)E);
    // 3) invert counts -> 1/deg (0 for isolated nodes)
    invert_kernel<<<(N_NODES + 255) / 256, 256, 0, stream>>>(cnt, N_NODES);

    // 4) scatter x into agg1 (f32 hardware atomics, resolves in L2)
    scatter48_kernel<<<8192, 256, 0, stream>>>(x, src, dst, agg1, E);

    // 5) layer 1: fused mean + dual GEMM + bias + ReLU  (WMMA f32 16x16x4)
    {
        int waves = (N_NODES / 16) * (F1 / 16);          // 25000
        int blocks = (waves * 32 + 255) / 256;           // 3125
        sage_layer1_kernel<<<blocks, 256, 0, stream>>>(x, agg1, cnt, W1l, W1r, b1, h);
    }

    // 6) scatter h into agg2
    scatter64_kernel<<<8192, 256, 0, stream>>>(h, src, dst, agg2, E);

    // 7) layer 2: fused mean + dual GEMM + bias  (WMMA f32 16x16x4)
    {
        int waves = (N_NODES / 16) * (F2 / 16);          // 12500
        int blocks = (waves * 32 + 255) / 256;           // 1563
        sage_layer2_kernel<<<blocks, 256, 0, stream>>>(h, agg2, cnt, W2l, W2r, b2, out);
    }
}